// stan_model_9380208574593
// MI455X (gfx1250) — compile-verified
//
#include <hip/hip_runtime.h>
#include <hip/hip_bf16.h>
#include <math.h>

// ---------------------------------------------------------------------------
// MI455X (gfx1250) implementation.
// Dominant cost: fc1 GEMM (256 x 197568 x 1024), bandwidth-bound on lw1
// (809 MB fp32) -> lw1 is read exactly once (block owns an N=64 x K-chunk
// slab, loops all 16 M-tiles reusing B fragments). fp32 WMMA
// (v_wmma_f32_16x16x4_f32) keeps reference precision. Split-K=16 with a
// deterministic fixed-order reduction (no float atomics -> bitwise
// reproducible across graph replays).
//
// Workspace layout (floats):
//   z       : 256*197568          = 50,577,408   (conv output, fc1 A matrix)
//   partial : 16*256*1024         =  4,194,304   (split-K partials)
//   h1      : 256*1024
//   h2      : 256*256
//   h3      : 256*32
// total ~220.5 MB.
// ---------------------------------------------------------------------------

typedef float v2f __attribute__((ext_vector_type(2)));
typedef float v8f __attribute__((ext_vector_type(8)));

#define B_SZ   256
#define T_SZ   8
#define S_SZ   8
#define F_SZ   16
#define H_SZ   64
#define F2C    32
#define F4C    64
#define NFILT  64
#define FLATK  197568      /* 64*7*7*63 */
#define N1     1024
#define N2     256
#define N3     32
#define KSPLIT 16
#define KCHUNK 12348       /* FLATK/KSPLIT, multiple of 4 */

// ---------------------------------------------------------------------------
// Kernel A: per-batch-element fused attention (stage 1 over T, stage 2 over S),
// X4 assembly (incl. the reference's (s-1,t)->(t,s-1) reshape quirk), and the
// 2x2x2 conv + bias + relu, writing z row-major (B, FLATK).
// ---------------------------------------------------------------------------
__global__ __launch_bounds__(256) void attn_conv_kernel(
    const float* __restrict__ X,  const float* __restrict__ aW1,
    const float* __restrict__ aU1, const float* __restrict__ aV1,
    const float* __restrict__ aW2, const float* __restrict__ aU2,
    const float* __restrict__ aV2, const float* __restrict__ convw,
    const float* __restrict__ convb, float* __restrict__ z) {
  __shared__ float sX [T_SZ*S_SZ*F_SZ];   // 1024
  __shared__ float sW [T_SZ*S_SZ*H_SZ];   // 4096 (Wx, then Wx2, then X4)
  __shared__ float sU [T_SZ*S_SZ*H_SZ];   // 4096 (Ux, then Ux2)
  __shared__ float sA1[T_SZ*T_SZ*S_SZ];   // 512  e1/a1 laid out [i][j][s]
  __shared__ float sA2[T_SZ*S_SZ*S_SZ];   // 512  e2/a2 laid out [t][i][j]
  __shared__ float sX2[T_SZ*S_SZ*F2C];    // 2048
  __shared__ float sCw[NFILT*8];          // 512
  __shared__ float sCb[NFILT];            // 64

  const int b   = blockIdx.x;
  const int tid = threadIdx.x;

  // phase 1: stage inputs
  for (int i = tid; i < 1024; i += 256) sX[i] = X[(size_t)b*1024 + i];
  for (int i = tid; i < 512;  i += 256) sCw[i] = convw[i];
  if (tid < 64) sCb[tid] = convb[tid];
  __syncthreads();

  // phase 2: Wx = X@aW1, Ux = X@aU1   (64 rows, K=16, H=64)
  for (int t2 = tid; t2 < 8192; t2 += 256) {
    int which = t2 >> 12;
    int idx   = t2 & 4095;
    int row = idx >> 6, h = idx & 63;
    const float* Wm = which ? aU1 : aW1;
    float s = 0.f;
#pragma unroll
    for (int f = 0; f < 16; ++f) s += sX[row*16 + f] * Wm[f*64 + h];
    (which ? sU : sW)[idx] = s;
  }
  __syncthreads();

  // phase 3: e1[i][j][s] = sum_h tanh(Wx[i,s,h] + Ux[j,s,h]) * aV1[h]
  for (int idx = tid; idx < 512; idx += 256) {
    int i = idx >> 6, j = (idx >> 3) & 7, s = idx & 7;
    const float* wr = &sW[(i*8 + s)*64];
    const float* ur = &sU[(j*8 + s)*64];
    float e = 0.f;
    for (int h = 0; h < 64; ++h) e += tanhf(wr[h] + ur[h]) * aV1[h];
    sA1[idx] = e;
  }
  __syncthreads();

  // phase 4: softmax over j (incl. diag), then zero the diagonal
  if (tid < 64) {
    int i = tid >> 3, s = tid & 7;
    float m = -1e30f;
    for (int j = 0; j < 8; ++j) m = fmaxf(m, sA1[i*64 + j*8 + s]);
    float ex[8]; float den = 0.f;
    for (int j = 0; j < 8; ++j) { ex[j] = expf(sA1[i*64 + j*8 + s] - m); den += ex[j]; }
    for (int j = 0; j < 8; ++j) sA1[i*64 + j*8 + s] = (j == i) ? 0.f : ex[j] / den;
  }
  __syncthreads();

  // phase 5: c1[i,s,f] = sum_j a1[i,j,s]*X[j,s,f];  X2 = concat(X, c1)
  for (int idx = tid; idx < 1024; idx += 256) {
    int i = idx >> 7, s = (idx >> 4) & 7, f = idx & 15;
    float c = 0.f;
    for (int j = 0; j < 8; ++j) c += sA1[i*64 + j*8 + s] * sX[(j*8 + s)*16 + f];
    sX2[(i*8 + s)*32 + f]      = sX[(i*8 + s)*16 + f];
    sX2[(i*8 + s)*32 + 16 + f] = c;
  }
  __syncthreads();

  // phase 6: Wx2 = X2@aW2, Ux2 = X2@aU2  (64 rows, K=32, H=64), reuse sW/sU
  for (int t2 = tid; t2 < 8192; t2 += 256) {
    int which = t2 >> 12;
    int idx   = t2 & 4095;
    int row = idx >> 6, h = idx & 63;
    const float* Wm = which ? aU2 : aW2;
    float s = 0.f;
#pragma unroll
    for (int f = 0; f < 32; ++f) s += sX2[row*32 + f] * Wm[f*64 + h];
    (which ? sU : sW)[idx] = s;
  }
  __syncthreads();

  // phase 7: e2[t][i][j]
  for (int idx = tid; idx < 512; idx += 256) {
    int t = idx >> 6, i = (idx >> 3) & 7, j = idx & 7;
    const float* wr = &sW[(t*8 + i)*64];
    const float* ur = &sU[(t*8 + j)*64];
    float e = 0.f;
    for (int h = 0; h < 64; ++h) e += tanhf(wr[h] + ur[h]) * aV2[h];
    sA2[idx] = e;
  }
  __syncthreads();

  // phase 8: softmax over j
  if (tid < 64) {
    int t = tid >> 3, i = tid & 7;
    float m = -1e30f;
    for (int j = 0; j < 8; ++j) m = fmaxf(m, sA2[t*64 + i*8 + j]);
    float ex[8]; float den = 0.f;
    for (int j = 0; j < 8; ++j) { ex[j] = expf(sA2[t*64 + i*8 + j] - m); den += ex[j]; }
    for (int j = 0; j < 8; ++j) sA2[t*64 + i*8 + j] = ex[j] / den;
  }
  __syncthreads();

  // phase 9: c2 with the reference's (s-1)*t flatten/regroup quirk; X4 in sW
  float* sX4 = sW;   // sW (Wx2) is dead now; 8*8*64 == 4096 floats fits exactly
  for (int idx = tid; idx < 2048; idx += 256) {
    int t = idx >> 8, i = (idx >> 5) & 7, f = idx & 31;
    float c = 0.f;
#pragma unroll
    for (int k = 0; k < 7; ++k) {
      int m  = t*7 + k;          // flat index regrouped as (t_new, k)
      int k2 = m >> 3;           // original off-diag slot
      int to = m & 7;            // original t
      int joff = (k2 < i) ? k2 : k2 + 1;   // offdiag[i][k2]
      c += sA2[to*64 + i*8 + joff] * sX2[(to*8 + joff)*32 + f];
    }
    sX4[(t*8 + i)*64 + f]      = sX2[(t*8 + i)*32 + f];
    sX4[(t*8 + i)*64 + 32 + f] = c;
  }
  __syncthreads();

  // phase 10: conv 2x2x2 (C_in=1, 64 filters) + bias + relu -> z (row-major)
  for (int p = tid; p < 3087; p += 256) {   // 7*7*63 output positions
    int ow  = p % 63;
    int rem = p / 63;
    int oh  = rem % 7;
    int od  = rem / 7;
    float xv[8];
#pragma unroll
    for (int kd = 0; kd < 2; ++kd)
#pragma unroll
      for (int kh = 0; kh < 2; ++kh)
#pragma unroll
        for (int kw = 0; kw < 2; ++kw)
          xv[kd*4 + kh*2 + kw] = sX4[((od+kd)*8 + (oh+kh))*64 + (ow+kw)];
    for (int oc = 0; oc < 64; ++oc) {
      float acc = sCb[oc];
#pragma unroll
      for (int q = 0; q < 8; ++q) acc += xv[q] * sCw[oc*8 + q];
      z[(size_t)b*FLATK + (size_t)oc*3087 + od*441 + oh*63 + ow] = fmaxf(acc, 0.f);
    }
  }
}

// ---------------------------------------------------------------------------
// fc1: split-K WMMA GEMM. Block = 8 waves handles (all M=256) x (N=64 slab)
// over one K-chunk. Each wave: one 16-wide N column, 8 M-tiles -> B fragment
// reused 8x, lw1 read exactly once overall.
// ---------------------------------------------------------------------------
__global__ __launch_bounds__(256) void fc1_partial_kernel(
    const float* __restrict__ zA, const float* __restrict__ lw1,
    float* __restrict__ partial) {
  const int w    = threadIdx.x >> 5;
  const int lane = threadIdx.x & 31;
  const int half = lane >> 4;
  const int l15  = lane & 15;
  const int n0     = blockIdx.x * 64 + (w & 3) * 16;
  const int mtBase = (w >> 2);              // 0 or 1
  const int kBeg = blockIdx.y * KCHUNK;
  const int kEnd = kBeg + KCHUNK;
  const int col  = n0 + l15;

  v8f acc[8];
  v8f vz = {0.f,0.f,0.f,0.f,0.f,0.f,0.f,0.f};
#pragma unroll
  for (int i = 0; i < 8; ++i) acc[i] = vz;

  for (int k = kBeg; k < kEnd; k += 4) {
    const int kk = k + 2*half;              // A/B K-pair per lane half
    v2f bf;
    bf.x = lw1[(size_t)kk * N1 + col];
    bf.y = lw1[(size_t)(kk + 1) * N1 + col];
    __builtin_prefetch(lw1 + (size_t)(kk + 16) * N1 + col, 0, 1);
#pragma unroll
    for (int i = 0; i < 8; ++i) {
      const int row = (mtBase + 2*i) * 16 + l15;
      v2f af = *(const v2f*)(zA + (size_t)row * FLATK + kk);
      acc[i] = __builtin_amdgcn_wmma_f32_16x16x4_f32(
          false, af, false, bf, (short)0, acc[i], false, false);
    }
  }

  float* slab = partial + (size_t)blockIdx.y * (B_SZ * N1);
#pragma unroll
  for (int i = 0; i < 8; ++i) {
    const int mt = mtBase + 2*i;
#pragma unroll
    for (int r = 0; r < 8; ++r) {
      const int row = mt*16 + r + half*8;   // C/D layout: vgpr r -> M=r / r+8
      slab[(size_t)row * N1 + col] = acc[i][r];
    }
  }
}

// Deterministic fixed-order split-K reduction + bias + relu.
__global__ __launch_bounds__(256) void fc1_reduce_kernel(
    const float* __restrict__ partial, const float* __restrict__ lb1,
    float* __restrict__ h1) {
  int idx = blockIdx.x * 256 + threadIdx.x;
  if (idx >= B_SZ * N1) return;
  float s = lb1[idx & (N1 - 1)];
#pragma unroll
  for (int ks = 0; ks < KSPLIT; ++ks)
    s += partial[(size_t)ks * (B_SZ * N1) + idx];
  h1[idx] = fmaxf(s, 0.f);
}

// ---------------------------------------------------------------------------
// Generic small WMMA GEMM: C = act(A[MxK] @ B[KxN] + bias). One wave per
// 16x16 output tile, full-K loop of v_wmma_f32_16x16x4_f32.
// ---------------------------------------------------------------------------
template<int M, int N, int K, bool RELU>
__global__ __launch_bounds__(256) void gemm_bias_act_kernel(
    const float* __restrict__ A, const float* __restrict__ Bm,
    const float* __restrict__ bias, float* __restrict__ C) {
  const int id = blockIdx.x * 8 + (threadIdx.x >> 5);
  constexpr int NT = N / 16;
  const int mt = id / NT, nt = id % NT;
  if (mt >= M / 16) return;                 // wave-uniform
  const int lane = threadIdx.x & 31;
  const int half = lane >> 4;
  const int l15  = lane & 15;
  const int m0   = mt * 16;
  const int col  = nt * 16 + l15;

  v8f acc = {0.f,0.f,0.f,0.f,0.f,0.f,0.f,0.f};
  for (int k = 0; k < K; k += 4) {
    const int kk = k + 2*half;
    v2f af = *(const v2f*)(A + (size_t)(m0 + l15) * K + kk);
    v2f bf;
    bf.x = Bm[(size_t)kk * N + col];
    bf.y = Bm[(size_t)(kk + 1) * N + col];
    acc = __builtin_amdgcn_wmma_f32_16x16x4_f32(
        false, af, false, bf, (short)0, acc, false, false);
  }
  const float bv = bias[col];
#pragma unroll
  for (int r = 0; r < 8; ++r) {
    const int row = m0 + r + half*8;
    float v = acc[r] + bv;
    if (RELU) v = fmaxf(v, 0.f);
    C[(size_t)row * N + col] = v;
  }
}

// fc4: 256x32 @ 32x2 + bias (tiny; scalar)
__global__ __launch_bounds__(512) void fc4_kernel(
    const float* __restrict__ h3, const float* __restrict__ lw4,
    const float* __restrict__ lb4, float* __restrict__ out) {
  int idx = blockIdx.x * blockDim.x + threadIdx.x;
  if (idx >= B_SZ * 2) return;
  int m = idx >> 1, c = idx & 1;
  float s = lb4[c];
#pragma unroll
  for (int k = 0; k < 32; ++k) s += h3[m*32 + k] * lw4[k*2 + c];
  out[m*2 + c] = s;
}

// ---------------------------------------------------------------------------
extern "C" void kernel_launch(void* const* d_in, const int* in_sizes, int n_in,
                              void* d_out, int out_size, void* d_ws, size_t ws_size,
                              hipStream_t stream) {
  (void)in_sizes; (void)n_in; (void)out_size; (void)ws_size;
  const float* X   = (const float*)d_in[0];
  const float* aW1 = (const float*)d_in[1];
  const float* aU1 = (const float*)d_in[2];
  const float* aV1 = (const float*)d_in[3];
  const float* aW2 = (const float*)d_in[4];
  const float* aU2 = (const float*)d_in[5];
  const float* aV2 = (const float*)d_in[6];
  const float* cw  = (const float*)d_in[7];
  const float* cb  = (const float*)d_in[8];
  const float* lw1 = (const float*)d_in[9];
  const float* lb1 = (const float*)d_in[10];
  const float* lw2 = (const float*)d_in[11];
  const float* lb2 = (const float*)d_in[12];
  const float* lw3 = (const float*)d_in[13];
  const float* lb3 = (const float*)d_in[14];
  const float* lw4 = (const float*)d_in[15];
  const float* lb4 = (const float*)d_in[16];
  float* out = (float*)d_out;

  float* W       = (float*)d_ws;
  float* z       = W;
  float* partial = z       + (size_t)B_SZ * FLATK;
  float* h1      = partial + (size_t)KSPLIT * B_SZ * N1;
  float* h2      = h1      + (size_t)B_SZ * N1;
  float* h3      = h2      + (size_t)B_SZ * N2;

  attn_conv_kernel<<<B_SZ, 256, 0, stream>>>(X, aW1, aU1, aV1, aW2, aU2, aV2,
                                             cw, cb, z);
  fc1_partial_kernel<<<dim3(16, KSPLIT), 256, 0, stream>>>(z, lw1, partial);
  fc1_reduce_kernel<<<(B_SZ * N1) / 256, 256, 0, stream>>>(partial, lb1, h1);
  gemm_bias_act_kernel<256, 256, 1024, true><<<32, 256, 0, stream>>>(h1, lw2, lb2, h2);
  gemm_bias_act_kernel<256, 32, 256,  true><<<4,  256, 0, stream>>>(h2, lw3, lb3, h3);
  fc4_kernel<<<1, 512, 0, stream>>>(h3, lw4, lb4, out);
}